// TemporalFusionTransformer_19868518711669
// MI455X (gfx1250) — compile-verified
//
#include <hip/hip_runtime.h>
#include <hip/hip_bf16.h>

#define B_   64
#define T_   512
#define V_   32
#define D_   64
#define NH_  4
#define HD_  16
#define LH_  128
#define G4_  512
#define FC_  64
#define EPS_ 1e-5f

typedef float v8f __attribute__((ext_vector_type(8)));
typedef float v2f __attribute__((ext_vector_type(2)));

__device__ __forceinline__ v8f wmma4(v2f a, v2f b, v8f c) {
  // D = A(16x4,f32) x B(4x16,f32) + C(16x16,f32)   -> v_wmma_f32_16x16x4_f32
  return __builtin_amdgcn_wmma_f32_16x16x4_f32(false, a, false, b, (short)0, c, false, false);
}

__device__ __forceinline__ float sigm(float x) { return 1.0f / (1.0f + __expf(-x)); }
__device__ __forceinline__ float eluf(float x) { return x > 0.f ? x : (__expf(x) - 1.f); }

// ---------------------------------------------------------------------------
// Generic WMMA GEMM: C(MxN) = A(MxK) @ B(KxN) + bias ; one wave per 16x16 tile
// ---------------------------------------------------------------------------
__global__ void gemm_bias(const float* __restrict__ A, const float* __restrict__ Bm,
                          const float* __restrict__ bias, float* __restrict__ C,
                          int M, int N, int K) {
  int lane = threadIdx.x & 31;
  int wave = blockIdx.x * (blockDim.x >> 5) + (threadIdx.x >> 5);
  int tilesN = N >> 4;
  int tm = (wave / tilesN) << 4;
  int tn = (wave % tilesN) << 4;
  if (tm >= M) return;
  int ml = lane & 15, kh = lane >> 4;
  v8f acc = {};
  for (int kb = 0; kb < K; kb += 4) {
    v2f a, b;
    const float* Ar = A + (size_t)(tm + ml) * K + kb + 2 * kh;
    a.x = Ar[0];
    a.y = Ar[1];
    const float* Br = Bm + (size_t)(kb + 2 * kh) * N + tn + ml;
    b.x = Br[0];
    b.y = Br[N];
    acc = wmma4(a, b, acc);
  }
  for (int r = 0; r < 8; ++r) {
    int row = tm + r + 8 * kh, col = tn + ml;
    float bv = bias ? bias[col] : 0.f;
    C[(size_t)row * N + col] = acc[r] + bv;
  }
}

// ---------------------------------------------------------------------------
// Fused Variable Selection Network: 16 tokens per block (256 thr = 8 waves).
// Per-token y(V,D) tile lives in LDS; never touches HBM.
// ---------------------------------------------------------------------------
__global__ void vsn_fused(
    const float* __restrict__ x,
    const float* __restrict__ Wi, const float* __restrict__ bi,
    const float* __restrict__ Ws, const float* __restrict__ bs,
    const float* __restrict__ W2, const float* __restrict__ b2,
    const float* __restrict__ Wg, const float* __restrict__ bg,
    const float* __restrict__ lng, const float* __restrict__ lnb,
    const float* __restrict__ gInW, const float* __restrict__ gInb,
    const float* __restrict__ gL2W, const float* __restrict__ gL2b,
    const float* __restrict__ gGluW, const float* __restrict__ gGlub,
    const float* __restrict__ gLng, const float* __restrict__ gLnb,
    const float* __restrict__ gSkW, const float* __restrict__ gSkb,
    float* __restrict__ sel) {
  extern __shared__ float sm[];
  float* yS  = sm;              // 16*2048  (y tile, f32)
  float* hS  = yS + 16 * 2048;  // 8*1024   per-wave scratch
  float* h2S = hS + 8 * 1024;   // 8*1024
  float* xS  = h2S + 8 * 1024;  // 16*32
  float* pS  = xS + 512;        // 8*256    K-split partials

  int tid = threadIdx.x;
  int lane = tid & 31;
  int w = tid >> 5;
  int ml = lane & 15;
  int khl = lane >> 4;
  int tb = blockIdx.x * 16;

  for (int i = tid; i < 512; i += 256) xS[i] = x[(size_t)tb * V_ + i];
  __syncthreads();

  float* hW = hS + w * 1024;
  float* h2W = h2S + w * 1024;

  // ---- per-variable transform, v strided over 8 waves ----
  for (int v = w; v < V_; v += 8) {
    for (int i = lane; i < 1024; i += 32) {
      int tok = i >> 6, d = i & 63;
      hW[i] = eluf(xS[tok * V_ + v] * Wi[v * D_ + d] + bi[v * D_ + d]);
    }
    asm volatile("s_wait_dscnt 0" ::: "memory");
    const float* W2v = W2 + (size_t)v * D_ * D_;
    for (int nt = 0; nt < 4; ++nt) {
      v8f acc = {};
      for (int kb = 0; kb < 64; kb += 4) {
        v2f a, bb;
        a.x = hW[ml * 64 + kb + 2 * khl];
        a.y = hW[ml * 64 + kb + 2 * khl + 1];
        bb.x = W2v[(kb + 2 * khl) * 64 + nt * 16 + ml];
        bb.y = W2v[(kb + 2 * khl + 1) * 64 + nt * 16 + ml];
        acc = wmma4(a, bb, acc);
      }
      for (int r = 0; r < 8; ++r) {
        int row = r + 8 * khl, col = nt * 16 + ml;
        h2W[row * 64 + col] = acc[r] + b2[v * D_ + col];
      }
    }
    asm volatile("s_wait_dscnt 0" ::: "memory");
    const float* Wgv = Wg + (size_t)v * D_ * 2 * D_;
    for (int nt = 0; nt < 4; ++nt) {
      v8f accA = {}, accB = {};
      for (int kb = 0; kb < 64; kb += 4) {
        v2f a, bA, bB;
        a.x = h2W[ml * 64 + kb + 2 * khl];
        a.y = h2W[ml * 64 + kb + 2 * khl + 1];
        bA.x = Wgv[(kb + 2 * khl) * 128 + nt * 16 + ml];
        bA.y = Wgv[(kb + 2 * khl + 1) * 128 + nt * 16 + ml];
        bB.x = Wgv[(kb + 2 * khl) * 128 + 64 + nt * 16 + ml];
        bB.y = Wgv[(kb + 2 * khl + 1) * 128 + 64 + nt * 16 + ml];
        accA = wmma4(a, bA, accA);
        accB = wmma4(a, bB, accB);
      }
      for (int r = 0; r < 8; ++r) {
        int row = r + 8 * khl, col = nt * 16 + ml;
        float av = accA[r] + bg[v * 128 + col];
        float bv = accB[r] + bg[v * 128 + 64 + col];
        float res = xS[row * V_ + v] * Ws[v * D_ + col] + bs[v * D_ + col];
        yS[row * 2048 + v * 64 + col] = av * sigm(bv) + res;
      }
    }
    asm volatile("s_wait_dscnt 0" ::: "memory");
    if (lane < 16) {  // per-(tok,v) layernorm over D
      float* yr = yS + lane * 2048 + v * 64;
      float mean = 0.f;
      for (int d = 0; d < 64; ++d) mean += yr[d];
      mean *= (1.f / 64.f);
      float var = 0.f;
      for (int d = 0; d < 64; ++d) { float u = yr[d] - mean; var += u * u; }
      float rs = rsqrtf(var * (1.f / 64.f) + EPS_);
      for (int d = 0; d < 64; ++d)
        yr[d] = (yr[d] - mean) * rs * lng[v * D_ + d] + lnb[v * D_ + d];
    }
    asm volatile("s_wait_dscnt 0" ::: "memory");
  }
  __syncthreads();

  // ---- weight GRN on flat (16 x 2048) ----
  float* h1S = hS;           // 16*64
  float* gS = hS + 1024;     // 16*64
  float* resS = hS + 2048;   // 16*32
  float* wS = hS + 2560;     // 16*32

  {  // h1 = flat @ inW ; K split 2 ways x N split 4 across 8 waves
    int nt = w & 3, ks = w >> 2;
    v8f acc = {};
    for (int kb = ks * 1024; kb < ks * 1024 + 1024; kb += 4) {
      v2f a, bb;
      a.x = yS[ml * 2048 + kb + 2 * khl];
      a.y = yS[ml * 2048 + kb + 2 * khl + 1];
      bb.x = gInW[(size_t)(kb + 2 * khl) * 64 + nt * 16 + ml];
      bb.y = gInW[(size_t)(kb + 2 * khl + 1) * 64 + nt * 16 + ml];
      acc = wmma4(a, bb, acc);
    }
    for (int r = 0; r < 8; ++r) pS[w * 256 + (r + 8 * khl) * 16 + ml] = acc[r];
  }
  __syncthreads();
  for (int i = tid; i < 1024; i += 256) {
    int tok = i >> 6, c = i & 63;
    int nt = c >> 4, cc = c & 15;
    h1S[i] = eluf(pS[nt * 256 + tok * 16 + cc] + pS[(nt + 4) * 256 + tok * 16 + cc] + gInb[c]);
  }
  __syncthreads();
  if (w < 4) {  // h2 = h1 @ l2W + b
    int nt = w;
    v8f acc = {};
    for (int kb = 0; kb < 64; kb += 4) {
      v2f a, bb;
      a.x = h1S[ml * 64 + kb + 2 * khl];
      a.y = h1S[ml * 64 + kb + 2 * khl + 1];
      bb.x = gL2W[(kb + 2 * khl) * 64 + nt * 16 + ml];
      bb.y = gL2W[(kb + 2 * khl + 1) * 64 + nt * 16 + ml];
      acc = wmma4(a, bb, acc);
    }
    for (int r = 0; r < 8; ++r)
      h2S[(r + 8 * khl) * 64 + nt * 16 + ml] = acc[r] + gL2b[nt * 16 + ml];
  }
  __syncthreads();
  if (w < 4) {  // g = h2 @ gluW + b  (2*dout = 64 cols)
    int nt = w;
    v8f acc = {};
    for (int kb = 0; kb < 64; kb += 4) {
      v2f a, bb;
      a.x = h2S[ml * 64 + kb + 2 * khl];
      a.y = h2S[ml * 64 + kb + 2 * khl + 1];
      bb.x = gGluW[(kb + 2 * khl) * 64 + nt * 16 + ml];
      bb.y = gGluW[(kb + 2 * khl + 1) * 64 + nt * 16 + ml];
      acc = wmma4(a, bb, acc);
    }
    for (int r = 0; r < 8; ++r)
      gS[(r + 8 * khl) * 64 + nt * 16 + ml] = acc[r] + gGlub[nt * 16 + ml];
  }
  __syncthreads();
  {  // res = flat @ skipW ; N split 2 x K split 4
    int nt = w & 1, ks = w >> 1;
    v8f acc = {};
    for (int kb = ks * 512; kb < ks * 512 + 512; kb += 4) {
      v2f a, bb;
      a.x = yS[ml * 2048 + kb + 2 * khl];
      a.y = yS[ml * 2048 + kb + 2 * khl + 1];
      bb.x = gSkW[(size_t)(kb + 2 * khl) * 32 + nt * 16 + ml];
      bb.y = gSkW[(size_t)(kb + 2 * khl + 1) * 32 + nt * 16 + ml];
      acc = wmma4(a, bb, acc);
    }
    for (int r = 0; r < 8; ++r) pS[w * 256 + (r + 8 * khl) * 16 + ml] = acc[r];
  }
  __syncthreads();
  for (int i = tid; i < 512; i += 256) {
    int tok = i >> 5, c = i & 31;
    int nt = c >> 4, cc = c & 15;
    float s = gSkb[c];
    for (int qq = 0; qq < 4; ++qq) s += pS[(nt + 2 * qq) * 256 + tok * 16 + cc];
    resS[i] = s;
  }
  __syncthreads();
  if (tid < 16) {  // glu + LN + softmax over V
    int tok = tid;
    float e[32];
    float mean = 0.f;
    for (int vv = 0; vv < 32; ++vv) {
      float val = gS[tok * 64 + vv] * sigm(gS[tok * 64 + 32 + vv]) + resS[tok * 32 + vv];
      e[vv] = val;
      mean += val;
    }
    mean *= (1.f / 32.f);
    float var = 0.f;
    for (int vv = 0; vv < 32; ++vv) { float u = e[vv] - mean; var += u * u; }
    float rs = rsqrtf(var * (1.f / 32.f) + EPS_);
    float mx = -1e30f;
    for (int vv = 0; vv < 32; ++vv) {
      e[vv] = (e[vv] - mean) * rs * gLng[vv] + gLnb[vv];
      mx = fmaxf(mx, e[vv]);
    }
    float sum = 0.f;
    for (int vv = 0; vv < 32; ++vv) { e[vv] = __expf(e[vv] - mx); sum += e[vv]; }
    float inv = 1.f / sum;
    for (int vv = 0; vv < 32; ++vv) wS[tok * 32 + vv] = e[vv] * inv;
  }
  __syncthreads();
  for (int i = tid; i < 1024; i += 256) {  // sel = sum_v y*w
    int tok = i >> 6, d = i & 63;
    float s = 0.f;
    for (int v = 0; v < V_; ++v) s += yS[tok * 2048 + v * 64 + d] * wS[tok * 32 + v];
    sel[(size_t)(tb + tok) * D_ + d] = s;
  }
}

// ---------------------------------------------------------------------------
// Sequential LSTM: block = 16 batch rows (128 thr = 4 waves).
// Whole Whh (128x512 f32 = 256KB) stays resident in the 320KB WGP LDS.
// G = precomputed x@Wih + b for all (b,t).
// ---------------------------------------------------------------------------
__global__ void lstm_seq(const float* __restrict__ G, const float* __restrict__ Whh,
                         float* __restrict__ H) {
  extern __shared__ float sm[];
  float* whhS = sm;             // 128*512
  float* hS = whhS + 65536;     // 16*128
  float* cS = hS + 2048;        // 16*128
  float* gS = cS + 2048;        // 16*512
  int tid = threadIdx.x, lane = tid & 31, w = tid >> 5;
  int b0 = blockIdx.x * 16;
  int ml = lane & 15, khl = lane >> 4;
  for (int i = tid; i < 65536; i += 128) whhS[i] = Whh[i];
  for (int i = tid; i < 2048; i += 128) { hS[i] = 0.f; cS[i] = 0.f; }
  __syncthreads();
  for (int t = 0; t < T_; ++t) {
    for (int nt = w * 8; nt < w * 8 + 8; ++nt) {
      v8f acc = {};
      for (int kb = 0; kb < 128; kb += 4) {
        v2f a, b;
        a.x = hS[ml * 128 + kb + 2 * khl];
        a.y = hS[ml * 128 + kb + 2 * khl + 1];
        b.x = whhS[(kb + 2 * khl) * 512 + nt * 16 + ml];
        b.y = whhS[(kb + 2 * khl + 1) * 512 + nt * 16 + ml];
        acc = wmma4(a, b, acc);
      }
      for (int r = 0; r < 8; ++r) {
        int row = r + 8 * khl, col = nt * 16 + ml;
        gS[row * 512 + col] = acc[r] + G[((size_t)(b0 + row) * T_ + t) * 512 + col];
      }
    }
    __syncthreads();
    for (int i = tid; i < 2048; i += 128) {
      int row = i >> 7, d = i & 127;
      float ig = gS[row * 512 + d], fg = gS[row * 512 + 128 + d];
      float gg = gS[row * 512 + 256 + d], og = gS[row * 512 + 384 + d];
      float c = sigm(fg) * cS[i] + sigm(ig) * tanhf(gg);
      float h = sigm(og) * tanhf(c);
      cS[i] = c;
      hS[i] = h;
      H[((size_t)(b0 + row) * T_ + t) * LH_ + d] = h;
    }
    __syncthreads();
  }
}

// ---------------------------------------------------------------------------
// Fused GRN (din=hid=dout=64), 16 rows per block (128 thr = 4 waves)
// ---------------------------------------------------------------------------
__global__ void grn64(const float* __restrict__ x, const float* __restrict__ inW,
                      const float* __restrict__ inb, const float* __restrict__ l2W,
                      const float* __restrict__ l2b, const float* __restrict__ gluW,
                      const float* __restrict__ glub, const float* __restrict__ lng,
                      const float* __restrict__ lnb, float* __restrict__ out) {
  __shared__ float h1S[1024], h2S[1024], gS[2048];
  int tid = threadIdx.x, lane = tid & 31, w = tid >> 5;
  int tb = blockIdx.x * 16;
  int ml = lane & 15, khl = lane >> 4;
  {
    v8f acc = {};
    for (int kb = 0; kb < 64; kb += 4) {
      v2f a, b;
      a.x = x[(size_t)(tb + ml) * 64 + kb + 2 * khl];
      a.y = x[(size_t)(tb + ml) * 64 + kb + 2 * khl + 1];
      b.x = inW[(kb + 2 * khl) * 64 + w * 16 + ml];
      b.y = inW[(kb + 2 * khl + 1) * 64 + w * 16 + ml];
      acc = wmma4(a, b, acc);
    }
    for (int r = 0; r < 8; ++r) {
      int row = r + 8 * khl, col = w * 16 + ml;
      h1S[row * 64 + col] = eluf(acc[r] + inb[col]);
    }
  }
  __syncthreads();
  {
    v8f acc = {};
    for (int kb = 0; kb < 64; kb += 4) {
      v2f a, b;
      a.x = h1S[ml * 64 + kb + 2 * khl];
      a.y = h1S[ml * 64 + kb + 2 * khl + 1];
      b.x = l2W[(kb + 2 * khl) * 64 + w * 16 + ml];
      b.y = l2W[(kb + 2 * khl + 1) * 64 + w * 16 + ml];
      acc = wmma4(a, b, acc);
    }
    for (int r = 0; r < 8; ++r) {
      int row = r + 8 * khl, col = w * 16 + ml;
      h2S[row * 64 + col] = acc[r] + l2b[col];
    }
  }
  __syncthreads();
  for (int half = 0; half < 2; ++half) {
    v8f acc = {};
    int n0 = half * 64 + w * 16;
    for (int kb = 0; kb < 64; kb += 4) {
      v2f a, b;
      a.x = h2S[ml * 64 + kb + 2 * khl];
      a.y = h2S[ml * 64 + kb + 2 * khl + 1];
      b.x = gluW[(kb + 2 * khl) * 128 + n0 + ml];
      b.y = gluW[(kb + 2 * khl + 1) * 128 + n0 + ml];
      acc = wmma4(a, b, acc);
    }
    for (int r = 0; r < 8; ++r) {
      int row = r + 8 * khl, col = n0 + ml;
      gS[row * 128 + col] = acc[r] + glub[col];
    }
  }
  __syncthreads();
  if (tid < 16) {
    int row = tid;
    float y[64];
    float mean = 0.f;
    for (int d = 0; d < 64; ++d) {
      float val = gS[row * 128 + d] * sigm(gS[row * 128 + 64 + d]) + x[(size_t)(tb + row) * 64 + d];
      y[d] = val;
      mean += val;
    }
    mean *= (1.f / 64.f);
    float var = 0.f;
    for (int d = 0; d < 64; ++d) { float u = y[d] - mean; var += u * u; }
    float rs = rsqrtf(var * (1.f / 64.f) + EPS_);
    for (int d = 0; d < 64; ++d)
      out[(size_t)(tb + row) * 64 + d] = (y[d] - mean) * rs * lng[d] + lnb[d];
  }
}

// ---------------------------------------------------------------------------
// LayerNorm(a + r) over dim 64, one wave per row
// ---------------------------------------------------------------------------
__global__ void ln_res64(const float* __restrict__ a, const float* __restrict__ r,
                         const float* __restrict__ g, const float* __restrict__ b,
                         float* __restrict__ out, int rows) {
  int wave = blockIdx.x * (blockDim.x >> 5) + (threadIdx.x >> 5);
  int lane = threadIdx.x & 31;
  if (wave >= rows) return;
  float e0 = a[(size_t)wave * 64 + lane] + r[(size_t)wave * 64 + lane];
  float e1 = a[(size_t)wave * 64 + 32 + lane] + r[(size_t)wave * 64 + 32 + lane];
  float s = e0 + e1;
  for (int o = 16; o > 0; o >>= 1) s += __shfl_xor(s, o, 32);
  float mean = s * (1.f / 64.f);
  float u0 = e0 - mean, u1 = e1 - mean;
  float v = u0 * u0 + u1 * u1;
  for (int o = 16; o > 0; o >>= 1) v += __shfl_xor(v, o, 32);
  float rs = rsqrtf(v * (1.f / 64.f) + EPS_);
  out[(size_t)wave * 64 + lane] = u0 * rs * g[lane] + b[lane];
  out[(size_t)wave * 64 + 32 + lane] = u1 * rs * g[32 + lane] + b[32 + lane];
}

// ---------------------------------------------------------------------------
// Flash attention: one wave per (b, head, q-tile of 16). Causal, online softmax.
// Q,K packed (B,T,D) with head h at cols [16h,16h+16); V is (B,T,HD) shared.
// ---------------------------------------------------------------------------
__global__ void attn_flash(const float* __restrict__ Q, const float* __restrict__ Kt,
                           const float* __restrict__ Vv, float* __restrict__ O) {
  __shared__ float sT[8][256];
  __shared__ float rowS[8][48];
  int lane = threadIdx.x & 31, w = threadIdx.x >> 5;
  int wave = blockIdx.x * 8 + w;
  int qt = wave & 31;
  int h = (wave >> 5) & 3;
  int b = wave >> 7;
  float* S = sT[w];
  float* mrow = rowS[w];
  float* lrow = mrow + 16;
  float* frow = mrow + 32;
  int ml = lane & 15, khl = lane >> 4;
  v2f qa[4];
  const float* Qb = Q + ((size_t)b * T_ + qt * 16) * D_ + h * 16;
  for (int kc = 0; kc < 4; ++kc) {
    qa[kc].x = Qb[ml * D_ + kc * 4 + 2 * khl];
    qa[kc].y = Qb[ml * D_ + kc * 4 + 2 * khl + 1];
  }
  if (lane < 16) { mrow[lane] = -1e30f; lrow[lane] = 0.f; }
  asm volatile("s_wait_dscnt 0" ::: "memory");
  v8f accO = {};
  const float scale = 0.25f;  // HD^-0.5
  for (int kt = 0; kt <= qt; ++kt) {
    v8f s = {};
    const float* Kb = Kt + ((size_t)b * T_ + kt * 16) * D_ + h * 16;
    for (int kc = 0; kc < 4; ++kc) {
      v2f bf;  // B[d][n] = K[key n][d] (transpose via addressing)
      bf.x = Kb[ml * D_ + kc * 4 + 2 * khl];
      bf.y = Kb[ml * D_ + kc * 4 + 2 * khl + 1];
      s = wmma4(qa[kc], bf, s);
    }
    for (int r = 0; r < 8; ++r) {
      int row = r + 8 * khl, col = ml;
      float val = s[r] * scale;
      if (kt * 16 + col > qt * 16 + row) val = -1e9f;
      S[row * 16 + col] = val;
    }
    asm volatile("s_wait_dscnt 0" ::: "memory");
    if (lane < 16) {
      int row = lane;
      float mx = mrow[row];
      for (int j = 0; j < 16; ++j) mx = fmaxf(mx, S[row * 16 + j]);
      float f = __expf(mrow[row] - mx);
      float sum = 0.f;
      for (int j = 0; j < 16; ++j) {
        float p = __expf(S[row * 16 + j] - mx);
        S[row * 16 + j] = p;
        sum += p;
      }
      lrow[row] = lrow[row] * f + sum;
      mrow[row] = mx;
      frow[row] = f;
    }
    asm volatile("s_wait_dscnt 0" ::: "memory");
    for (int r = 0; r < 8; ++r) accO[r] *= frow[r + 8 * khl];
    const float* Vb = Vv + ((size_t)b * T_ + kt * 16) * HD_;
    for (int kc = 0; kc < 4; ++kc) {
      v2f a, bf;
      a.x = S[ml * 16 + kc * 4 + 2 * khl];
      a.y = S[ml * 16 + kc * 4 + 2 * khl + 1];
      bf.x = Vb[(kc * 4 + 2 * khl) * HD_ + ml];
      bf.y = Vb[(kc * 4 + 2 * khl + 1) * HD_ + ml];
      accO = wmma4(a, bf, accO);
    }
  }
  asm volatile("s_wait_dscnt 0" ::: "memory");
  float* Ob = O + (((size_t)b * NH_ + h) * T_ + qt * 16) * HD_;
  for (int r = 0; r < 8; ++r) {
    int row = r + 8 * khl, col = ml;
    Ob[row * HD_ + col] = accO[r] / lrow[row];
  }
}

__global__ void mean_heads(const float* __restrict__ O, float* __restrict__ Vm) {
  int i = blockIdx.x * blockDim.x + threadIdx.x;
  if (i >= B_ * T_ * HD_) return;
  int b = i / (T_ * HD_);
  int rem = i % (T_ * HD_);
  float s = 0.f;
  for (int h = 0; h < NH_; ++h) s += O[((size_t)b * NH_ + h) * T_ * HD_ + rem];
  Vm[i] = s * 0.25f;
}

__global__ void head_kernel(const float* __restrict__ fin, const float* __restrict__ W1,
                            const float* __restrict__ b1, const float* __restrict__ bng,
                            const float* __restrict__ bnb, const float* __restrict__ W2,
                            const float* __restrict__ b2, float* __restrict__ out) {
  int b = threadIdx.x;
  if (b >= B_) return;
  const float* xr = fin + ((size_t)b * T_ + (T_ - 1)) * D_;
  float s = rsqrtf(1.0f + EPS_);
  float acc = b2[0];
  for (int j = 0; j < FC_; ++j) {
    float h = b1[j];
    for (int d = 0; d < D_; ++d) h += xr[d] * W1[d * FC_ + j];
    h = h * s * bng[j] + bnb[j];
    h = 0.5f * h * (1.0f + erff(h * 0.70710678118654752f));  // exact gelu
    acc += h * W2[j];
  }
  out[b] = acc;
}

// ---------------------------------------------------------------------------
extern "C" void kernel_launch(void* const* d_in, const int* in_sizes, int n_in,
                              void* d_out, int out_size, void* d_ws, size_t ws_size,
                              hipStream_t stream) {
  (void)in_sizes; (void)n_in; (void)out_size; (void)ws_size;
  auto F = [&](int i) { return (const float*)d_in[i]; };
  // Input order (setup_inputs insertion order, depth-first):
  // 0:x  vsn: 1:Wi 2:bi 3:Ws 4:bs 5:W2 6:b2 7:Wg 8:bg 9:ln_g 10:ln_b
  // wgrn: 11:inW 12:inb 13:l2W 14:l2b 15:gluW 16:glub 17:lng 18:lnb 19:skipW 20:skipb
  // lstm: 21:Wih1 22:Whh1 23:b1 24:Wih2 25:Whh2 26:b2
  // 27:projW 28:projb 29:lstm_lng 30:lstm_lnb
  // post_lstm_grn: 31..38 ; 39:qW 40:qb 41:kW 42:kb 43:vW 44:vb 45:oW 46:ob
  // attn_grn: 47..54 ; 55:attn_lng 56:attn_lnb ; final_grn: 57..64 ; 65:fin_lng 66:fin_lnb
  // 67:fc1W 68:fc1b 69:bn_g 70:bn_b 71:fc2W 72:fc2b
  float* ws = (float*)d_ws;
  const int BT = B_ * T_;
  size_t o = 0;
  float* sel = ws + o;   o += (size_t)BT * D_;
  float* G = ws + o;     o += (size_t)BT * G4_;
  float* H1 = ws + o;    o += (size_t)BT * LH_;
  float* H2 = ws + o;    o += (size_t)BT * LH_;
  float* tmp = ws + o;   o += (size_t)BT * D_;
  float* hln = ws + o;   o += (size_t)BT * D_;
  float* gated = ws + o; o += (size_t)BT * D_;
  float* q = ws + o;     o += (size_t)BT * D_;
  float* k = ws + o;     o += (size_t)BT * D_;
  float* vv = ws + o;    o += (size_t)BT * HD_;
  float* O = ws + o;     o += (size_t)BT * D_;  // B*NH*T*HD
  float* Vm = ws + o;    o += (size_t)BT * HD_;
  float* attn = ws + o;  o += (size_t)BT * D_;
  float* a1 = ws + o;    o += (size_t)BT * D_;
  float* a2 = ws + o;    o += (size_t)BT * D_;
  float* f1 = ws + o;    o += (size_t)BT * D_;
  float* fin = ws + o;   o += (size_t)BT * D_;

  auto gemm = [&](const float* A, const float* Bm, const float* bias, float* C, int M, int N,
                  int K) {
    int waves = (M / 16) * (N / 16);
    int blocks = (waves * 32 + 255) / 256;
    gemm_bias<<<blocks, 256, 0, stream>>>(A, Bm, bias, C, M, N, K);
  };

  // 1) fused VSN -> sel (B,T,D)
  vsn_fused<<<BT / 16, 256, 206848, stream>>>(
      F(0), F(1), F(2), F(3), F(4), F(5), F(6), F(7), F(8), F(9), F(10), F(11), F(12), F(13),
      F(14), F(15), F(16), F(17), F(18), F(19), F(20), sel);

  // 2) LSTM layer 1: input GEMM hoisted, then sequential recurrence (Whh in LDS)
  gemm(sel, F(21), F(23), G, BT, G4_, D_);
  lstm_seq<<<B_ / 16, 128, 311296, stream>>>(G, F(22), H1);
  // 3) LSTM layer 2
  gemm(H1, F(24), F(26), G, BT, G4_, LH_);
  lstm_seq<<<B_ / 16, 128, 311296, stream>>>(G, F(25), H2);

  // 4) proj + LN residual vs sel
  gemm(H2, F(27), F(28), tmp, BT, D_, LH_);
  ln_res64<<<BT / 8, 256, 0, stream>>>(tmp, sel, F(29), F(30), hln, BT);

  // 5) post-LSTM GRN
  grn64<<<BT / 16, 128, 0, stream>>>(hln, F(31), F(32), F(33), F(34), F(35), F(36), F(37), F(38),
                                     gated);

  // 6) QKV projections
  gemm(gated, F(39), F(40), q, BT, D_, D_);
  gemm(gated, F(41), F(42), k, BT, D_, D_);
  gemm(gated, F(43), F(44), vv, BT, HD_, D_);

  // 7) causal flash attention per head, then head-average + output proj
  attn_flash<<<(B_ * NH_ * (T_ / 16)) / 8, 256, 0, stream>>>(q, k, vv, O);
  mean_heads<<<(BT * HD_) / 256, 256, 0, stream>>>(O, Vm);
  gemm(Vm, F(45), F(46), attn, BT, D_, HD_);

  // 8) attn GRN + LN residual vs gated
  grn64<<<BT / 16, 128, 0, stream>>>(attn, F(47), F(48), F(49), F(50), F(51), F(52), F(53), F(54),
                                     a1);
  ln_res64<<<BT / 8, 256, 0, stream>>>(a1, gated, F(55), F(56), a2, BT);

  // 9) final GRN + LN residual
  grn64<<<BT / 16, 128, 0, stream>>>(a2, F(57), F(58), F(59), F(60), F(61), F(62), F(63), F(64),
                                     f1);
  ln_res64<<<BT / 8, 256, 0, stream>>>(f1, a2, F(65), F(66), fin, BT);

  // 10) last-token MLP head -> (B,)
  head_kernel<<<1, 64, 0, stream>>>(fin, F(67), F(68), F(69), F(70), F(71), F(72), (float*)d_out);
}